// MultiBCE_22806276342355
// MI455X (gfx1250) — compile-verified
//
#include <hip/hip_runtime.h>

// Problem constants (from reference): B=512, H=8, C=8192, output = scalar f32.
#define B_DIM 512
#define H_DIM 8
#define C_DIM 8192
#define BLOCK 256
#define GRID_MAIN 1024   // 1024 blocks * 256 thr * 4 iters * 4 elems = 512*8192

typedef __attribute__((ext_vector_type(2))) float v2f;
typedef __attribute__((ext_vector_type(8))) float v8f;

// Full 32-lane wave sum via V_WMMA_F32_16X16X4_F32.
// A (16x4 f32): per-lane VGPR0 = acc, VGPR1 = 0. ISA layout: lanes 0-15 supply
// K=0/K=1, lanes 16-31 supply K=2/K=3 for rows M=lane%16. B = all ones (4x16).
// D[m][n] = acc[m] + acc[m+16] for every n. Lane<16 holds rows 0..7 in its 8
// D-VGPRs, lane>=16 holds rows 8..15 -> sum own 8 + two shuffles = wave total.
__device__ __forceinline__ float wave_sum_wmma(float x) {
  v2f a; a[0] = x;    a[1] = 0.0f;
  v2f b; b[0] = 1.0f; b[1] = 1.0f;
  v8f c = {};
  v8f d = __builtin_amdgcn_wmma_f32_16x16x4_f32(
      /*neg_a=*/false, a, /*neg_b=*/false, b,
      /*c_mod=*/(short)0, c, /*reuse_a=*/false, /*reuse_b=*/false);
  float s = d[0] + d[1] + d[2] + d[3] + d[4] + d[5] + d[6] + d[7];
  return __shfl(s, 0, 32) + __shfl(s, 16, 32);
}

// torch-style BCE element with log clamp at -100, specialized for binary t:
//   -(t*log(p) + (1-t)*log1p(-p)) == -max(log(t ? p : 1-p), -100)
// Single hardware v_log_f32 per element; p=0 -> log(0)=-inf -> clamp -100.
__device__ __forceinline__ float bce_elem(float p, float t) {
  float sel = (t != 0.0f) ? p : (1.0f - p);
  return -fmaxf(__logf(sel), -100.0f);
}

// w[c] = sum_h lam[h] * La[h,c]  (collapses the H dimension analytically)
__global__ void mbce_prep_w(const float* __restrict__ La,
                            const float* __restrict__ lam,
                            float* __restrict__ w) {
  int c = blockIdx.x * blockDim.x + threadIdx.x;
  if (c < C_DIM) {
    float s = 0.0f;
#pragma unroll
    for (int h = 0; h < H_DIM; ++h)
      s = fmaf(lam[h], La[h * C_DIM + c], s);
    w[c] = s;
  }
}

// Streaming weighted BCE reduction: partial[block] = sum over this block's
// slice of w[c] * bce(y_pred, y_true). Deterministic (fixed-order trees).
__global__ void __launch_bounds__(BLOCK)
mbce_main(const float4* __restrict__ yp4,
          const float4* __restrict__ yt4,
          const float4* __restrict__ w4,
          float* __restrict__ partial) {
  const int tid    = blockIdx.x * BLOCK + threadIdx.x;
  const int stride = GRID_MAIN * BLOCK;
  float acc = 0.0f;
#pragma unroll
  for (int k = 0; k < 4; ++k) {
    int i = tid + k * stride;
    if (k < 3) {  // prefetch next tile of the stream (global_prefetch_b8)
      __builtin_prefetch(&yp4[i + stride], 0, 1);
      __builtin_prefetch(&yt4[i + stride], 0, 1);
    }
    float4 p  = yp4[i];
    float4 t  = yt4[i];
    float4 wv = w4[i & (C_DIM / 4 - 1)];  // c-block index = i mod (C/4)
    acc = fmaf(wv.x, bce_elem(p.x, t.x), acc);
    acc = fmaf(wv.y, bce_elem(p.y, t.y), acc);
    acc = fmaf(wv.z, bce_elem(p.z, t.z), acc);
    acc = fmaf(wv.w, bce_elem(p.w, t.w), acc);
  }
  float ws = wave_sum_wmma(acc);
  __shared__ float swave[BLOCK / 32];
  int lane = threadIdx.x & 31, wid = threadIdx.x >> 5;
  if (lane == 0) swave[wid] = ws;
  __syncthreads();
  if (threadIdx.x == 0) {
    float bs = 0.0f;
#pragma unroll
    for (int i = 0; i < BLOCK / 32; ++i) bs += swave[i];
    partial[blockIdx.x] = bs;
  }
}

// Single-block deterministic combine of the GRID_MAIN partials; scale by 1/C.
__global__ void __launch_bounds__(BLOCK)
mbce_final(const float* __restrict__ partial, float* __restrict__ out) {
  float acc = 0.0f;
#pragma unroll
  for (int k = 0; k < GRID_MAIN / BLOCK; ++k)
    acc += partial[threadIdx.x + k * BLOCK];
  float ws = wave_sum_wmma(acc);
  __shared__ float swave[BLOCK / 32];
  int lane = threadIdx.x & 31, wid = threadIdx.x >> 5;
  if (lane == 0) swave[wid] = ws;
  __syncthreads();
  if (threadIdx.x == 0) {
    float s = 0.0f;
#pragma unroll
    for (int i = 0; i < BLOCK / 32; ++i) s += swave[i];
    out[0] = s * (1.0f / (float)C_DIM);
  }
}

extern "C" void kernel_launch(void* const* d_in, const int* in_sizes, int n_in,
                              void* d_out, int out_size, void* d_ws, size_t ws_size,
                              hipStream_t stream) {
  const float* y_pred = (const float*)d_in[0];  // [512, 8192]
  const float* y_true = (const float*)d_in[1];  // [512, 8192]
  const float* La     = (const float*)d_in[2];  // [8, 8192]
  const float* lam    = (const float*)d_in[3];  // [8]

  float* w       = (float*)d_ws;        // 8192 floats
  float* partial = w + C_DIM;           // GRID_MAIN floats
  float* out     = (float*)d_out;       // scalar

  mbce_prep_w<<<C_DIM / BLOCK, BLOCK, 0, stream>>>(La, lam, w);
  mbce_main<<<GRID_MAIN, BLOCK, 0, stream>>>(
      (const float4*)y_pred, (const float4*)y_true, (const float4*)w, partial);
  mbce_final<<<1, BLOCK, 0, stream>>>(partial, out);
}